// Dilation1D_9474697855598
// MI455X (gfx1250) — compile-verified
//
#include <hip/hip_runtime.h>
#include <stdint.h>
#include <math.h>

// Dilation1D: out[b,i] = max_{j in [0,200]} x[b, i+j-100] + h[j],
//   h[j] = -(j-100)^2 / (4*scale), out-of-range x treated as -inf.
// Tropical (max-plus) op => no WMMA path exists; we use the CDNA5
// async global->LDS pipeline (ASYNCcnt) + VALU add/max3 out of LDS.

#define RPB   8      // rows per block (one wave32 per row)
#define LROW  201    // row length
#define XPLEN 404    // padded row in LDS: valid x in [100,300], -inf elsewhere
#define TOUT  7      // outputs per lane (32*7 = 224 >= 201)
#define WREG  10     // rolling window = TOUT + 3 (4 offsets per iteration)

__device__ __forceinline__ uint32_t lds_off(const void* p) {
  // Flat shared-aperture address: low 32 bits are the LDS byte offset.
  return (uint32_t)(uintptr_t)p;
}

__global__ __launch_bounds__(256) void dilate1d_kernel(
    const float* __restrict__ x,
    const float* __restrict__ scale,
    float* __restrict__ out,
    int nrows)
{
  __shared__ __align__(16) float xp[RPB][XPLEN];
  __shared__ __align__(16) float htab[204];

  const int tid  = threadIdx.x;
  const int wave = tid >> 5;            // row within block
  const int lane = tid & 31;
  const int grow = blockIdx.x * RPB + wave;
  const float neg_inf = -__builtin_inff();

  // ---- 1) issue async global->LDS loads for THIS wave's row (ASYNCcnt) ----
  // Interior [100,300] is written only by the async engine; pads only by ds
  // stores below -> disjoint addresses, no ordering hazard.
  if (grow < nrows) {
    const float* base = x + (size_t)grow * LROW;  // uniform SGPR pair per wave
#pragma unroll
    for (int k = 0; k < 7; ++k) {
      int c = lane + k * 32;
      if (c < LROW) {
        uint32_t a    = lds_off(&xp[wave][100 + c]);
        uint32_t voff = (uint32_t)c * 4u;
        asm volatile(
          "global_load_async_to_lds_b32 %0, %1, %2 offset:0"
          :: "v"(a), "v"(voff), "s"(base) : "memory");
      }
    }
  }

  // ---- 2) fill -inf pads of own row (overlapped with async loads) ---------
#pragma unroll
  for (int k = 0; k < 4; ++k) {         // [0, 99]
    int p = lane + k * 32;
    if (p < 100) xp[wave][p] = neg_inf;
  }
#pragma unroll
  for (int k = 0; k < 4; ++k) {         // [301, 403]
    int p = 301 + lane + k * 32;
    if (p < XPLEN) xp[wave][p] = neg_inf;
  }

  // ---- 3) h table (single pass, shared by all waves) ----------------------
  if (tid < LROW) {
    const float s = scale[0];
    float d = (float)(tid - 100);
    htab[tid] = -(d * d) / (4.0f * s);
  }
  __syncthreads();                       // htab visible to all waves
  asm volatile("s_wait_asynccnt 0" ::: "memory");  // own row resident in LDS

  // ---- 4) compute: lane owns 7 contiguous outputs -------------------------
  const float* __restrict__ xrow = xp[wave];
  int ib = lane * TOUT;
  if (ib > LROW - TOUT) ib = LROW - TOUT;  // clamp: lanes 28..31 -> 194

  float xr[WREG];   // rolling window: xr[k] = xp_row[ib + k + oi]
  float acc[TOUT];
#pragma unroll
  for (int k = 0; k < WREG; ++k) xr[k] = xrow[ib + k];
#pragma unroll
  for (int t = 0; t < TOUT; ++t) acc[t] = neg_inf;

  // Main loop: 4 structuring-element offsets per iteration (oi = 0..199).
  for (int oi = 0; oi < 200; oi += 4) {
    const float4 hv = *reinterpret_cast<const float4*>(&htab[oi]); // ds_load_b128 broadcast
#pragma unroll
    for (int t = 0; t < TOUT; ++t) {
      float c0 = xr[t + 0] + hv.x;
      float c1 = xr[t + 1] + hv.y;
      float c2 = xr[t + 2] + hv.z;
      float c3 = xr[t + 3] + hv.w;
      acc[t] = fmaxf(fmaxf(acc[t], c0), c1);   // -> v_max3_num_f32
      acc[t] = fmaxf(fmaxf(acc[t], c2), c3);   // -> v_max3_num_f32
    }
#pragma unroll
    for (int k = 0; k < WREG - 4; ++k) xr[k] = xr[k + 4];
#pragma unroll
    for (int k = 0; k < 4; ++k) xr[WREG - 4 + k] = xrow[ib + oi + WREG + k];
  }
  // Remainder offset oi = 200.
  {
    const float hv = htab[200];
#pragma unroll
    for (int t = 0; t < TOUT; ++t) acc[t] = fmaxf(acc[t], xr[t] + hv);
  }

  // ---- 5) store (lanes 29..31 duplicate lane 28's clamped range; skip) ----
  if (grow < nrows && lane <= 28) {
    float* orow = out + (size_t)grow * LROW + ib;
#pragma unroll
    for (int t = 0; t < TOUT; ++t) orow[t] = acc[t];
  }
}

extern "C" void kernel_launch(void* const* d_in, const int* in_sizes, int n_in,
                              void* d_out, int out_size, void* d_ws, size_t ws_size,
                              hipStream_t stream) {
  const float* x     = (const float*)d_in[0];
  const float* scale = (const float*)d_in[1];
  float* out         = (float*)d_out;
  const int nrows  = in_sizes[0] / LROW;                 // 131072
  const int blocks = (nrows + RPB - 1) / RPB;            // 16384
  hipLaunchKernelGGL(dilate1d_kernel, dim3(blocks), dim3(256), 0, stream,
                     x, scale, out, nrows);
}